// SamplingTargetLayer_66778151518378
// MI455X (gfx1250) — compile-verified
//
#include <hip/hip_runtime.h>
#include <hip/hip_bf16.h>
#include <math.h>

// Problem constants (match reference setup_inputs)
#define BB 4
#define RR 1024
#define NN 100
#define WAVES_PER_BLK 8
#define THREADS 256

// Flat element offsets into d_out (all fields are 4 bytes: float32 or int32)
#define OUT_ROIS 0
#define OUT_GT   (BB * RR * 7)            // 28672
#define OUT_MAX  (OUT_GT + BB * RR * 8)   // 61440
#define OUT_LBL  (OUT_MAX + BB * RR)      // 65536
#define OUT_MASK (OUT_LBL + BB * RR)      // 69632

// Pointer/vector types for the CDNA5 async memory->LDS builtin
typedef __attribute__((ext_vector_type(4))) int v4i;
typedef __attribute__((address_space(1))) v4i* as1_v4i_ptr;
typedef __attribute__((address_space(3))) v4i* as3_v4i_ptr;

// Exact rotated-rect BEV intersection + 3D IoU for one (roi, gt) pair,
// mirroring the reference math (incl. stable argsort tie-breaking by index).
__device__ __forceinline__ float iou3d_pair(const float* __restrict__ A, float cA, float sA,
                                            const float* __restrict__ Bb, float cB, float sB)
{
    const float EPS = 1e-8f;

    // BEV corners (CCW), same corner order as reference
    float cax[4], cay[4], cbx[4], cby[4];
    {
        const float LX[4] = {0.5f, -0.5f, -0.5f, 0.5f};
        const float LY[4] = {0.5f, 0.5f, -0.5f, -0.5f};
#pragma unroll
        for (int i = 0; i < 4; i++) {
            float lx = LX[i] * A[3], ly = LY[i] * A[4];
            cax[i] = A[0] + lx * cA - ly * sA;
            cay[i] = A[1] + lx * sA + ly * cA;
            lx = LX[i] * Bb[3]; ly = LY[i] * Bb[4];
            cbx[i] = Bb[0] + lx * cB - ly * sB;
            cby[i] = Bb[1] + lx * sB + ly * cB;
        }
    }

    // 24 candidate vertices + validity; 8 sentinel slots for the 32-wide network.
    // vis[k] = (orig_index << 1) | valid  (index is the argsort tie-break key)
    float pxs[32], pys[32], axs[32];
    int   vis[32];

    // slots 0..3: corners of A tested against B
#pragma unroll
    for (int i = 0; i < 4; i++) {
        float dxp = cax[i] - Bb[0], dyp = cay[i] - Bb[1];
        float lx = dxp * cB + dyp * sB;
        float ly = -dxp * sB + dyp * cB;
        int v = (fabsf(lx) <= Bb[3] * 0.5f + 1e-5f) && (fabsf(ly) <= Bb[4] * 0.5f + 1e-5f);
        pxs[i] = cax[i]; pys[i] = cay[i]; vis[i] = (i << 1) | v;
    }
    // slots 4..7: corners of B tested against A
#pragma unroll
    for (int i = 0; i < 4; i++) {
        float dxp = cbx[i] - A[0], dyp = cby[i] - A[1];
        float lx = dxp * cA + dyp * sA;
        float ly = -dxp * sA + dyp * cA;
        int v = (fabsf(lx) <= A[3] * 0.5f + 1e-5f) && (fabsf(ly) <= A[4] * 0.5f + 1e-5f);
        pxs[4 + i] = cbx[i]; pys[4 + i] = cby[i]; vis[4 + i] = ((4 + i) << 1) | v;
    }
    // slots 8..23: edge-edge intersections (i-major, j-minor, same as reshape(16,2))
#pragma unroll
    for (int i = 0; i < 4; i++) {
        float a1x = cax[i], a1y = cay[i];
        float d1x = cax[(i + 1) & 3] - a1x, d1y = cay[(i + 1) & 3] - a1y;
#pragma unroll
        for (int j = 0; j < 4; j++) {
            float b1x = cbx[j], b1y = cby[j];
            float d2x = cbx[(j + 1) & 3] - b1x, d2y = cby[(j + 1) & 3] - b1y;
            float denom = d1x * d2y - d1y * d2x;
            float dfx = b1x - a1x, dfy = b1y - a1y;
            int nz = fabsf(denom) > EPS;
            float safe = nz ? denom : EPS;
            float t = (dfx * d2y - dfy * d2x) / safe;
            float u = (dfx * d1y - dfy * d1x) / safe;
            int hit = nz && (t >= 0.0f) && (t <= 1.0f) && (u >= 0.0f) && (u <= 1.0f);
            int k = 8 + i * 4 + j;
            pxs[k] = a1x + t * d1x;
            pys[k] = a1y + t * d1y;
            vis[k] = (k << 1) | hit;
        }
    }

    // vertex count + centroid over valid points (same accumulation order as reference)
    int nv = 0; float sx = 0.0f, sy = 0.0f;
#pragma unroll
    for (int k = 0; k < 24; k++)
        if (vis[k] & 1) { nv++; sx += pxs[k]; sy += pys[k]; }
    float inv = 1.0f / (float)(nv < 1 ? 1 : nv);
    float ctx = sx * inv, cty = sy * inv;

#pragma unroll
    for (int k = 0; k < 24; k++)
        axs[k] = (vis[k] & 1) ? atan2f(pys[k] - cty, pxs[k] - ctx) : 1e9f;
#pragma unroll
    for (int k = 24; k < 32; k++) { axs[k] = 2e9f; pxs[k] = 0.0f; pys[k] = 0.0f; vis[k] = (k << 1); }

    // Batcher odd-even merge sorting network, 32 slots, fully unrolled so every
    // array index is a compile-time constant (all-register, no scratch, no
    // wave divergence). Key = (angle, original index) => exactly reproduces
    // jnp.argsort's stable ordering.
#pragma unroll
    for (int p = 1; p < 32; p <<= 1) {
#pragma unroll
        for (int k = p; k >= 1; k >>= 1) {
#pragma unroll
            for (int j = k % p; j + k < 32; j += 2 * k) {
#pragma unroll
                for (int i = 0; i < k; i++) {
                    int lo = i + j, hi = i + j + k;
                    if (hi < 32 && (lo / (2 * p)) == (hi / (2 * p))) {
                        float aL = axs[lo], aH = axs[hi];
                        int iL = vis[lo], iH = vis[hi];
                        bool sw = (aH < aL) || ((aH == aL) && ((iH >> 1) < (iL >> 1)));
                        axs[lo] = sw ? aH : aL; axs[hi] = sw ? aL : aH;
                        vis[lo] = sw ? iH : iL; vis[hi] = sw ? iL : iH;
                        float pL = pxs[lo], pH = pxs[hi];
                        pxs[lo] = sw ? pH : pL; pxs[hi] = sw ? pL : pH;
                        pL = pys[lo]; pH = pys[hi];
                        pys[lo] = sw ? pH : pL; pys[hi] = sw ? pL : pH;
                    }
                }
            }
        }
    }

    // Shoelace over the first 24 sorted slots; invalid slots padded with the
    // first vertex (zero contribution), as in the reference.
    float fx0 = 0.0f, fy0 = 0.0f, prx = 0.0f, pry = 0.0f, acc = 0.0f;
#pragma unroll
    for (int m = 0; m < 24; m++) {
        float X = pxs[m], Y = pys[m];
        if (m == 0) { fx0 = X; fy0 = Y; }
        else {
            if (!(vis[m] & 1)) { X = fx0; Y = fy0; }
            acc += prx * Y - X * pry;
        }
        prx = X; pry = Y;
    }
    acc += prx * fy0 - fx0 * pry;
    float interBev = (nv >= 3) ? 0.5f * fabsf(acc) : 0.0f;

    // 3D part
    float topA = A[2] + A[5] * 0.5f, topB = Bb[2] + Bb[5] * 0.5f;
    float botA = A[2] - A[5] * 0.5f, botB = Bb[2] - Bb[5] * 0.5f;
    float oh = fmaxf(fminf(topA, topB) - fmaxf(botA, botB), 0.0f);
    float inter = interBev * oh;
    float va = A[3] * A[4] * A[5];
    float vb = Bb[3] * Bb[4] * Bb[5];
    return inter / fmaxf(va + vb - inter, 1e-6f);
}

// One wave32 per ROI; lanes stride over the N=100 GT boxes; butterfly-reduce
// (max, first-index) across the wave. GT boxes + their sin/cos staged in LDS
// once per block (reuse factor 1024 ROIs per batch slice).
__launch_bounds__(THREADS, 1)
__global__ void sampling_target_kernel(const float* __restrict__ rois,
                                       const int* __restrict__ labels,
                                       const float* __restrict__ gt,
                                       float* __restrict__ out)
{
    __shared__ float sGT[NN * 8];   // (N,8) gt boxes for this batch
    __shared__ float sCS[NN * 2];   // per-gt (cos, sin)

    const int tid  = threadIdx.x;
    const int wave = tid >> 5;
    const int lane = tid & 31;
    const int roiG = blockIdx.x * WAVES_PER_BLK + wave;  // global roi id in [0, B*R)
    const int b    = roiG / RR;                          // same for all waves in block

    const float* gbase = gt + b * NN * 8;

    // CDNA5 prefetch of this batch's gt slice (3200B = 25 x 128B)
    if (tid < 25) __builtin_prefetch(gbase + tid * 32, 0, 3);

    // ---- Stage gt boxes (200 x 16B = 3200B) into LDS ----
    // Preferred: CDNA5 async memory->LDS DMA path (no VGPR round-trip),
    // tracked by ASYNCcnt. Fallback: plain vectorized copy (known-good).
#if defined(__gfx1250__) && __has_builtin(__builtin_amdgcn_global_load_async_to_lds_b128)
    if (tid < NN * 2) {
        as1_v4i_ptr gsrc = (as1_v4i_ptr)(gbase + tid * 4);  // AS1, const cast away
        as3_v4i_ptr ldst = (as3_v4i_ptr)(sGT + tid * 4);    // AS3 (LDS)
        __builtin_amdgcn_global_load_async_to_lds_b128(gsrc, ldst, 0, 0);
    }
#else
    if (tid < NN * 2) ((float4*)sGT)[tid] = ((const float4*)gbase)[tid];
#endif

    // Per-gt sin/cos computed straight from global (independent of LDS staging)
    if (tid < NN) {
        float sh, ch;
        sincosf(gbase[tid * 8 + 6], &sh, &ch);
        sCS[2 * tid]     = ch;
        sCS[2 * tid + 1] = sh;
    }

#if defined(__gfx1250__) && __has_builtin(__builtin_amdgcn_global_load_async_to_lds_b128)
    // Each wave drains its own async LDS writes before the block barrier.
#if __has_builtin(__builtin_amdgcn_s_wait_asynccnt)
    __builtin_amdgcn_s_wait_asynccnt(0);
#else
    asm volatile("s_wait_asynccnt 0x0" ::: "memory");
#endif
#endif
    __syncthreads();

    // Load this wave's ROI box (broadcast load, all lanes same address)
    float A[7];
#pragma unroll
    for (int c = 0; c < 7; c++) A[c] = rois[roiG * 7 + c];
    float sA, cA;
    sincosf(A[6], &sA, &cA);

    // Each lane covers n = lane, lane+32, lane+64, lane+96 (ascending => first-max kept)
    float bestV = -1.0f;
    int   bestN = 0;
    for (int kk = 0; kk < (NN + 31) / 32; kk++) {
        int n = lane + kk * 32;
        if (n < NN) {
            float Bb[7];
#pragma unroll
            for (int c = 0; c < 7; c++) Bb[c] = sGT[n * 8 + c];
            float iou = iou3d_pair(A, cA, sA, Bb, sCS[2 * n], sCS[2 * n + 1]);
            if (iou > bestV) { bestV = iou; bestN = n; }
        }
    }

    // wave32 butterfly reduction: max value, ties -> lowest index (== jnp.argmax)
#pragma unroll
    for (int off = 16; off >= 1; off >>= 1) {
        float v2 = __shfl_xor(bestV, off, 32);
        int   i2 = __shfl_xor(bestN, off, 32);
        if (v2 > bestV || (v2 == bestV && i2 < bestN)) { bestV = v2; bestN = i2; }
    }

    // Emit all 5 outputs (every element written every call)
    if (lane < 8) {
        out[OUT_GT + roiG * 8 + lane] = sGT[bestN * 8 + lane];
    } else if (lane < 15) {
        int c = lane - 8;
        out[OUT_ROIS + roiG * 7 + c] = rois[roiG * 7 + c];
    } else if (lane == 15) {
        out[OUT_MAX + roiG] = bestV;
    } else if (lane == 16) {
        ((int*)out)[OUT_LBL + roiG] = labels[roiG];
    } else if (lane == 17) {
        ((int*)out)[OUT_MASK + roiG] = (bestV > 0.55f) ? 1 : 0;
    }
}

extern "C" void kernel_launch(void* const* d_in, const int* in_sizes, int n_in,
                              void* d_out, int out_size, void* d_ws, size_t ws_size,
                              hipStream_t stream) {
    (void)n_in; (void)out_size; (void)d_ws; (void)ws_size;
    const float* rois   = (const float*)d_in[0];  // (B,R,7) f32
    const int*   labels = (const int*)d_in[1];    // (B,R)   i32
    const float* gt     = (const float*)d_in[2];  // (B,N,8) f32
    float* out = (float*)d_out;

    int nRoi   = in_sizes[1];              // B*R = 4096
    int blocks = nRoi / WAVES_PER_BLK;     // 512 blocks x 256 threads (8 wave32 / block)
    sampling_target_kernel<<<blocks, THREADS, 0, stream>>>(rois, labels, gt, out);
}